// AVSL_Similarity_55482387529980
// MI455X (gfx1250) — compile-verified
//
#include <hip/hip_runtime.h>
#include <math.h>

typedef float v2f __attribute__((ext_vector_type(2)));
typedef float v8f __attribute__((ext_vector_type(8)));

// LDS layout (in floats)
#define HS   132                    // H row stride (pad: b64-aligned, conflict-free A loads)
#define WS   130                    // W row stride (pad: conflict-free B loads)
#define WMAT (128 * WS)             // one normalized link matrix
#define WN_OFF (256 * HS)           // 33792
#define NRM_OFF (WN_OFF + 2 * WMAT) // 67072
#define SMEM_FLOATS (NRM_OFF + 96)  // + 3 layers x (16 i + 16 j) inv-norms

struct Params {
    const float* in[20];
    float* out;
};

__global__ __launch_bounds__(256) void avsl_fused_kernel(Params p) {
    extern __shared__ float smem[];
    float* Hs  = smem;            // [256][HS]  running nodes_hat for 256 (i,j) pairs
    float* WN  = smem + WN_OFF;   // [2][128][WS] normalized links
    float* NRM = smem + NRM_OFF;  // [3][2][16]  row inverse L2 norms

    const int t      = threadIdx.x;
    const int w      = t >> 5;        // wave id (8 waves)
    const int lane   = t & 31;
    const int laneLo = lane & 15;
    const int laneHi = lane >> 4;
    const int i0     = blockIdx.y * 16;
    const int j0     = blockIdx.x * 16;

    // ---------- Phase A: top-3 mask + column-normalize both link matrices into LDS
    {
        const int m = t >> 7;         // which link matrix
        const int e = t & 127;        // column (topk along dim 0 = rows d)
        const float* L = p.in[18 + m];
        float v1 = -1e30f, v2 = -1e30f, v3 = -1e30f;
        for (int d = 0; d < 128; ++d) {
            float wv = L[d * 128 + e];
            if (wv > v1)      { v3 = v2; v2 = v1; v1 = wv; }
            else if (wv > v2) { v3 = v2; v2 = wv; }
            else if (wv > v3) { v3 = wv; }
        }
        const float inv = 1.0f / (v1 + v2 + v3 + 1e-8f);
        float* Wd = WN + m * WMAT;
        for (int d = 0; d < 128; ++d) {
            float wv = L[d * 128 + e];
            Wd[d * WS + e] = (wv >= v3) ? wv * inv : 0.0f;
        }
    }
    // ---------- Phase A2: row inverse L2 norms for the 16 i-rows / 16 j-rows, 3 layers
    if (t < 96) {
        const int l  = t / 32;
        const int rm = t & 31;
        const int s  = rm >> 4;       // 0 = emb1 row, 1 = emb2 row
        const int rr = rm & 15;
        const int row = (s == 0 ? i0 : j0) + rr;
        const float4* rp = (const float4*)(p.in[6 * l + s] + row * 128);
        float acc = 0.0f;
        for (int q = 0; q < 32; ++q) {
            float4 u = rp[q];
            acc += u.x * u.x + u.y * u.y + u.z * u.z + u.w * u.w;
        }
        NRM[l * 32 + s * 16 + rr] = 1.0f / fmaxf(sqrtf(acc), 1e-12f);
    }
    __syncthreads();

    // ---------- Phase B: H init = layer-0 nodes, thread t owns pair t = (i_local*16 + j_local)
    {
        const int il = t >> 4, jl = t & 15;
        const float inv_i = NRM[il];
        const float inv_j = NRM[16 + jl];
        const float4* r1 = (const float4*)(p.in[0] + (i0 + il) * 128);
        const float4* r2 = (const float4*)(p.in[1] + (j0 + jl) * 128);
        float4* hrow = (float4*)(Hs + t * HS);
        for (int q = 0; q < 32; ++q) {
            float4 a = r1[q], b = r2[q];
            float dx = a.x * inv_i - b.x * inv_j;
            float dy = a.y * inv_i - b.y * inv_j;
            float dz = a.z * inv_i - b.z * inv_j;
            float dw = a.w * inv_i - b.w * inv_j;
            float4 d;
            d.x = dx * dx; d.y = dy * dy; d.z = dz * dz; d.w = dw * dw;
            hrow[q] = d;
        }
    }
    __syncthreads();

    // Each wave owns pairs [32w, 32w+32): M = 2 tiles of 16 pairs, N = 8 tiles of 16 e.
    const int pairBase = w * 32;
    float outAcc[2][8];
#pragma unroll
    for (int mt = 0; mt < 2; ++mt)
#pragma unroll
        for (int r = 0; r < 8; ++r) outAcc[mt][r] = 0.0f;

    for (int l = 1; l <= 2; ++l) {
        const float* Wd = WN + (l - 1) * WMAT;

        // ---------- GEMM: T[pair,e] = sum_d H[pair,d] * W[d,e]  via V_WMMA_F32_16X16X4_F32
        v8f C[2][8];
#pragma unroll
        for (int mt = 0; mt < 2; ++mt)
#pragma unroll
            for (int nt = 0; nt < 8; ++nt) {
                v8f z = {0.f, 0.f, 0.f, 0.f, 0.f, 0.f, 0.f, 0.f};
                C[mt][nt] = z;
            }

        for (int kt = 0; kt < 32; ++kt) {
            const int k0 = kt * 4 + 2 * laneHi;   // this lane's K pair within the 16x16x4 step
            v2f a[2], b[8];
#pragma unroll
            for (int mt = 0; mt < 2; ++mt) {      // A: lane=M(pair), vgpr=K
                const float* ap = Hs + (pairBase + mt * 16 + laneLo) * HS + k0;
                a[mt][0] = ap[0];
                a[mt][1] = ap[1];
            }
#pragma unroll
            for (int nt = 0; nt < 8; ++nt) {      // B: lane=N(e), vgpr=K
                const float* bp = Wd + k0 * WS + nt * 16 + laneLo;
                b[nt][0] = bp[0];
                b[nt][1] = bp[WS];
            }
#pragma unroll
            for (int mt = 0; mt < 2; ++mt)
#pragma unroll
                for (int nt = 0; nt < 8; ++nt)
                    C[mt][nt] = __builtin_amdgcn_wmma_f32_16x16x4_f32(
                        false, a[mt], false, b[nt], (short)0, C[mt][nt], false, false);
        }

        // ---------- Fused epilogue: h' = T + P*(nodes_l - T), P = sigmoid(alpha*eta + beta)
        // C layout: value at (pair = pairBase + 16*mt + 8*laneHi + r, e = 16*nt + laneLo)
        const float* e1p = p.in[6 * l + 0];
        const float* e2p = p.in[6 * l + 1];
        const float* c1p = p.in[6 * l + 2];
        const float* c2p = p.in[6 * l + 3];
        const float* al  = p.in[6 * l + 4];
        const float* be  = p.in[6 * l + 5];
        const float* nrmL = NRM + l * 32;
        const int jlBase = 8 * laneHi;

#pragma unroll
        for (int nt = 0; nt < 8; ++nt) {
            const int e = nt * 16 + laneLo;
            const float aE = al[e], bE = be[e];
            float e1v[2], c1v[2];
#pragma unroll
            for (int mt = 0; mt < 2; ++mt) {
                const int il = 2 * w + mt;        // i_local is wave/mt constant
                e1v[mt] = e1p[(i0 + il) * 128 + e] * nrmL[il];
                c1v[mt] = c1p[(i0 + il) * 128 + e];
            }
#pragma unroll
            for (int r = 0; r < 8; ++r) {
                const int jl = jlBase + r;
                const float e2v = e2p[(j0 + jl) * 128 + e] * nrmL[16 + jl];
                const float c2v = c2p[(j0 + jl) * 128 + e];
#pragma unroll
                for (int mt = 0; mt < 2; ++mt) {
                    float diff  = e1v[mt] - e2v;
                    float nodes = diff * diff;
                    float x  = aE * (c1v[mt] * c2v) + bE;
                    float P  = 1.0f / (1.0f + __expf(-x));
                    float tv = C[mt][nt][r];
                    float hn = tv + P * (nodes - tv);
                    if (l < 2) {
                        // wave-private H rows -> safe writeback for next layer's GEMM
                        Hs[(pairBase + mt * 16 + jl) * HS + e] = hn;
                    } else {
                        outAcc[mt][r] += hn;      // final layer: reduce over e directly
                    }
                }
            }
        }
        __syncthreads();
    }

    // ---------- Final e-reduction: butterfly across the 16 lanes of each half-wave
#pragma unroll
    for (int mt = 0; mt < 2; ++mt)
#pragma unroll
        for (int r = 0; r < 8; ++r) {
            float v = outAcc[mt][r];
            v += __shfl_xor(v, 1, 32);
            v += __shfl_xor(v, 2, 32);
            v += __shfl_xor(v, 4, 32);
            v += __shfl_xor(v, 8, 32);
            if (laneLo == 0)
                p.out[(i0 + 2 * w + mt) * 512 + (j0 + 8 * laneHi + r)] = v;
        }
}

extern "C" void kernel_launch(void* const* d_in, const int* in_sizes, int n_in,
                              void* d_out, int out_size, void* d_ws, size_t ws_size,
                              hipStream_t stream) {
    (void)in_sizes; (void)n_in; (void)out_size; (void)d_ws; (void)ws_size;
    Params p;
    for (int i = 0; i < 20; ++i) p.in[i] = (const float*)d_in[i];
    p.out = (float*)d_out;
    dim3 grid(32, 32);   // 16x16 output tiles over [512,512]
    size_t shmem = SMEM_FLOATS * sizeof(float);  // ~262 KB of the 320 KB WGP LDS
    avsl_fused_kernel<<<grid, 256, shmem, stream>>>(p);
}